// PagedAttention_74612171866643
// MI455X (gfx1250) — compile-verified
//
#include <hip/hip_runtime.h>
#include <math.h>

// ---- problem constants (match reference) ----
static constexpr int kSeqs      = 32;
static constexpr int kHeads     = 16;
static constexpr int kHeadDim   = 128;
static constexpr int kBlockSz   = 16;   // tokens per cache page
static constexpr int kMaxBlocks = 128;  // pages per sequence
static constexpr int kWaves     = 4;    // waves per workgroup
#define ATTN_SCALE 0.08838834764831845f

typedef float v2f __attribute__((ext_vector_type(2)));
typedef float v8f __attribute__((ext_vector_type(8)));

// reduce over the 16-lane token group (masks 1,2,4,8); both 16-lane halves
// hold identical data so this is a full token reduction.
__device__ __forceinline__ float groupMax16(float v) {
#pragma unroll
  for (int m = 1; m <= 8; m <<= 1) v = fmaxf(v, __shfl_xor(v, m, 32));
  return v;
}
__device__ __forceinline__ float groupSum16(float v) {
#pragma unroll
  for (int m = 1; m <= 8; m <<= 1) v += __shfl_xor(v, m, 32);
  return v;
}

__global__ __launch_bounds__(kWaves * 32, 1)
void paged_attn_wmma_kernel(const float* __restrict__ q,
                            const float* __restrict__ kc,   // [NB][H][D/8][16][8]
                            const float* __restrict__ vc,   // [NB][H][128][16]
                            const int*   __restrict__ bt,   // [S][128]
                            const int*   __restrict__ ctx_lens,
                            float*       __restrict__ out)  // [S][H][128]
{
  const int h    = blockIdx.x;          // head
  const int s    = blockIdx.y;          // sequence
  const int tid  = threadIdx.x;         // 0..127
  const int w    = tid >> 5;            // wave 0..3
  const int lane = tid & 31;
  const int t    = lane & 15;           // token slot (QK N) / d-lane (AV N)
  const int hi   = lane >> 4;           // which 16-lane half
  const int hi2  = hi * 2;              // float offset of this half's K-pair

  __shared__ float sh_q[kHeadDim];            // pre-scaled query
  __shared__ float sh_o[kWaves][kHeadDim];    // per-wave partial output
  __shared__ float sh_m[kWaves];              // per-wave running max
  __shared__ float sh_s[kWaves];              // per-wave running sum

  // stage q * scale once per workgroup
  sh_q[tid] = q[((size_t)s * kHeads + h) * kHeadDim + tid] * ATTN_SCALE;
  __syncthreads();

  const int  ctx   = ctx_lens[s];
  const int  nblk  = (ctx + kBlockSz - 1) / kBlockSz;
  const int* btrow = bt + s * kMaxBlocks;
  const size_t headStride = (size_t)kHeadDim * kBlockSz;   // 2048 floats / page-head

  float m_run = -INFINITY;
  float s_run = 0.0f;
  v8f acc_av[8];
#pragma unroll
  for (int c = 0; c < 8; ++c)
#pragma unroll
    for (int e = 0; e < 8; ++e) acc_av[c][e] = 0.0f;

  // flash-decode loop: wave w owns pages w, w+4, w+8, ...
  for (int bi = w; bi < nblk; bi += kWaves) {
    const int phys = btrow[bi];
    // per-lane bases: fold in token slot and half-wave pair offset so the hot
    // loops issue exact b64 loads with only an immediate/scalar offset.
    const float* khl = kc + ((size_t)phys * kHeads + h) * headStride + t * 8  + hi2;
    const float* vhl = vc + ((size_t)phys * kHeads + h) * headStride + t * 16 + hi2;

    if (bi + kWaves < nblk) {   // wave-uniform prefetch of next page
      const int pn = btrow[bi + kWaves];
      __builtin_prefetch(kc + ((size_t)pn * kHeads + h) * headStride, 0, 1);
      __builtin_prefetch(vc + ((size_t)pn * kHeads + h) * headStride, 0, 1);
    }

    // ---- QK^T: D(16x16) = Qrep(16x4) x K^T(4x16), 32 k-steps over D=128 ----
    v8f accqk;
#pragma unroll
    for (int e = 0; e < 8; ++e) accqk[e] = 0.0f;

#pragma unroll 4
    for (int i = 0; i < kHeadDim / 4; ++i) {
      const int db = i * 4;                  // d_base for this k-step
      // A: q replicated across M rows; lanes 0-15 hold k=0,1; lanes 16-31 k=2,3
      const float2 qa = *(const float2*)(sh_q + db + hi2);
      // B: K[tok=t, db+hi2 .. db+hi2+1]; paged offset = (d>>3)*128 + tok*8 + (d&7)
      const float2 kk = *(const float2*)(khl + (size_t)((db >> 3) * 128 + (db & 4)));
      v2f a; a.x = qa.x; a.y = qa.y;
      v2f b; b.x = kk.x; b.y = kk.y;
      accqk = __builtin_amdgcn_wmma_f32_16x16x4_f32(false, a, false, b,
                                                    (short)0, accqk, false, false);
    }

    // rows of D are identical -> every lane has logit for token (lane&15)
    float logit = accqk[0];
    const int tok = bi * kBlockSz + t;
    if (tok >= ctx) logit = -INFINITY;

    // ---- online softmax update (all quantities wave-uniform after reduce) ----
    const float mb    = groupMax16(logit);
    const float m_new = fmaxf(m_run, mb);          // finite: page has >=1 valid token
    const float corr  = __expf(m_run - m_new);     // 0 on first page
    const float p     = __expf(logit - m_new);     // 0 for masked tokens
    s_run = s_run * corr + groupSum16(p);
    m_run = m_new;
    if (corr != 1.0f) {                            // wave-uniform: s_cbranch skip
#pragma unroll
      for (int c = 0; c < 8; ++c)
#pragma unroll
        for (int e = 0; e < 8; ++e) acc_av[c][e] *= corr;
    }

    // ---- AV: out(16 d per chunk) += P(1x16) x V(16x128), K-dim = 4 tokens ----
#pragma unroll
    for (int tg = 0; tg < 4; ++tg) {
      const int t0 = tg * 4;
      // lanes 0-15 need p[t0],p[t0+1]; lanes 16-31 need p[t0+2],p[t0+3]
      v2f a;
      a.x = __shfl(p, t0 + hi2, 32);       // ds_bpermute (per-lane index)
      a.y = __shfl(p, t0 + hi2 + 1, 32);
#pragma unroll
      for (int c = 0; c < 8; ++c) {
        // V[t0+hi2 .. +1, d=c*16+t]; layout [d][tok], tokens contiguous
        const float2 vv = *(const float2*)(vhl + (size_t)(c * 256 + t0));
        v2f b; b.x = vv.x; b.y = vv.y;
        acc_av[c] = __builtin_amdgcn_wmma_f32_16x16x4_f32(false, a, false, b,
                                                          (short)0, acc_av[c], false, false);
      }
    }
  }

  // ---- cross-wave merge via LDS ----
  if (lane < 16) {
#pragma unroll
    for (int c = 0; c < 8; ++c) sh_o[w][c * 16 + t] = acc_av[c][0];
  }
  if (lane == 0) { sh_m[w] = m_run; sh_s[w] = s_run; }
  __syncthreads();

  const float M = fmaxf(fmaxf(sh_m[0], sh_m[1]), fmaxf(sh_m[2], sh_m[3]));
  float num = 0.0f, den = 0.0f;
#pragma unroll
  for (int wv = 0; wv < kWaves; ++wv) {
    const float f = __expf(sh_m[wv] - M);   // 0 for waves that saw no pages
    num += f * sh_o[wv][tid];
    den += f * sh_s[wv];
  }
  out[((size_t)s * kHeads + h) * kHeadDim + tid] = num / den;
}

extern "C" void kernel_launch(void* const* d_in, const int* in_sizes, int n_in,
                              void* d_out, int out_size, void* d_ws, size_t ws_size,
                              hipStream_t stream) {
  (void)in_sizes; (void)n_in; (void)out_size; (void)d_ws; (void)ws_size;
  const float* q   = (const float*)d_in[0];
  const float* kc  = (const float*)d_in[1];
  const float* vc  = (const float*)d_in[2];
  const int*   bt  = (const int*)d_in[3];
  const int*   cl  = (const int*)d_in[4];
  float*       out = (float*)d_out;

  dim3 grid(kHeads, kSeqs);           // 512 workgroups: one per (seq, head)
  dim3 block(kWaves * 32);            // 4 wave32 waves
  paged_attn_wmma_kernel<<<grid, block, 0, stream>>>(q, kc, vc, bt, cl, out);
}